// LNCCLoss_45715631899318
// MI455X (gfx1250) — compile-verified
//
#include <hip/hip_runtime.h>
#include <hip/hip_bf16.h>

typedef __attribute__((ext_vector_type(2))) float v2f;
typedef __attribute__((ext_vector_type(8))) float v8f;

// Problem geometry: [B=2, C=1, D=160, H=192, W=192] f32
#define kW      192
#define kH      192
#define kD      160
#define kPLANE  (kW * kH)          // 36864
#define kVOL    (kD * kPLANE)      // 5898240
#define kNVOX   (2 * kVOL)         // 11796480
#define TILE    16
#define HALO    20                 // TILE + 4 halo
#define ZCHUNK  20
#define NTX     (kW / TILE)        // 12
#define NTY     (kH / TILE)        // 12
#define NZC     (kD / ZCHUNK)      // 8
#define NTASK   (2 * NZC * NTY * NTX)  // 2304
#define RINGSZ  (5 * 6 * 256)      // 5 fields x 6 slots x 256 pixels

// One wave (32 lanes) per workgroup; each wave owns one 16x16xZCHUNK output
// tile. Per z-plane: stage 20x20 s/t into LDS, x-pass 5-tap sliding sums in
// registers (5 fields: s, t, s^2, t^2, s*t), y-pass as banded matmul
// P = R(16x20) x T(20x16) via 5x V_WMMA_F32_16X16X4_F32 per field (exact f32),
// z-pass as register running sums with a 6-deep LDS ring of P planes.
__global__ __launch_bounds__(32)
void lncc_tile_kernel(const float* __restrict__ src,
                      const float* __restrict__ tgt,
                      float* __restrict__ partial) {
  __shared__ float sS[HALO * HALO];        // staged source plane (halo coords)
  __shared__ float sT[HALO * HALO];        // staged target plane
  __shared__ float sTx[5 * HALO * TILE];   // x-summed fields: [f][k=0..19][n=0..15]
  __shared__ float sRing[RINGSZ];          // P-plane ring: [f][slot][j][lane]

  const int lane = threadIdx.x;            // 0..31
  const int half = lane >> 4;              // 0: lanes 0-15, 1: lanes 16-31
  const int mn   = lane & 15;              // M (A) / N (B,D) index for this lane

  int task = blockIdx.x;
  const int tx = task % NTX; task /= NTX;
  const int ty = task % NTY; task /= NTY;
  const int zc = task % NZC; task /= NZC;
  const int b  = task;                     // batch 0..1
  const int x0 = tx * TILE, y0 = ty * TILE, z0 = zc * ZCHUNK;
  const long base = (long)b * kVOL;

  // Zero the ring so warm-up iterations can subtract unconditionally
  // (read slot (zp+3)%6 is always either zero or written 5 iterations ago).
  for (int p = lane; p < RINGSZ; p += 32) sRing[p] = 0.0f;
  __syncthreads();

  // Constant A operand: R[m][k] = 1 iff m <= k <= m+4 (y-direction box sum).
  // f32 16x4 A layout: VGPR0 = K (4c+0 | 4c+2 per half), VGPR1 = K+1.
  v2f Areg[5];
#pragma unroll
  for (int c = 0; c < 5; ++c) {
    const int k0 = 4 * c + (half ? 2 : 0);
    const int k1 = k0 + 1;
    Areg[c].x = (k0 >= mn && k0 <= mn + 4) ? 1.0f : 0.0f;
    Areg[c].y = (k1 >= mn && k1 <= mn + 4) ? 1.0f : 0.0f;
  }

  // z-direction running window sums per field, per owned pixel slot (8/lane).
  float zs[5][8];
#pragma unroll
  for (int f = 0; f < 5; ++f)
#pragma unroll
    for (int j = 0; j < 8; ++j) zs[f][j] = 0.0f;

  float lacc = 0.0f;

  // Incremental mod-6 ring slots: at zp, write slot (zp+8)%6, read (zp+3)%6.
  int si = (z0 + 6) % 6;                   // (zp+8)%6 at zp = z0-2
  int so = (z0 + 1) % 6;                   // (zp+3)%6 at zp = z0-2

  for (int zp = z0 - 2; zp <= z0 + ZCHUNK + 1; ++zp) {   // 24 planes
    const bool pvalid = (zp >= 0) && (zp < kD);

    if (pvalid) {
      // ---- Stage s/t 20x20 (zero outside volume), prefetch next plane ----
      for (int p = lane; p < HALO * HALO; p += 32) {
        const int hy = p / HALO, hx = p - hy * HALO;
        const int gy = y0 + hy - 2, gx = x0 + hx - 2;
        float vs = 0.0f, vt = 0.0f;
        if ((unsigned)gy < (unsigned)kH && (unsigned)gx < (unsigned)kW) {
          const long idx = base + (long)zp * kPLANE + gy * kW + gx;
          vs = src[idx];
          vt = tgt[idx];
          if (zp + 1 < kD) {
            __builtin_prefetch(src + idx + kPLANE, 0, 1);  // global_prefetch_b8
            __builtin_prefetch(tgt + idx + kPLANE, 0, 1);
          }
        }
        sS[p] = vs;
        sT[p] = vt;
      }
      __syncthreads();

      // ---- x-pass: 5-tap sliding window over 5 fields, rows y=0..19 ----
      if (lane < HALO) {
        const int y = lane;
        float w0 = 0, w1 = 0, w2 = 0, w3 = 0, w4 = 0;
        float r0[5], r1[5], r2[5], r3[5], r4[5];
#pragma unroll
        for (int x = 0; x < HALO; ++x) {
          const int sl = x % 5;
          if (x >= 5) { w0 -= r0[sl]; w1 -= r1[sl]; w2 -= r2[sl]; w3 -= r3[sl]; w4 -= r4[sl]; }
          const float a = sS[y * HALO + x];
          const float t = sT[y * HALO + x];
          const float f2 = a * a, f3 = t * t, f4 = a * t;
          w0 += a; w1 += t; w2 += f2; w3 += f3; w4 += f4;
          r0[sl] = a; r1[sl] = t; r2[sl] = f2; r3[sl] = f3; r4[sl] = f4;
          if (x >= 4) {
            const int xo = x - 4;
            sTx[0 * 320 + y * 16 + xo] = w0;
            sTx[1 * 320 + y * 16 + xo] = w1;
            sTx[2 * 320 + y * 16 + xo] = w2;
            sTx[3 * 320 + y * 16 + xo] = w3;
            sTx[4 * 320 + y * 16 + xo] = w4;
          }
        }
      }
      __syncthreads();
    }

    // ---- y-pass via WMMA, then z running sums + ring update ----
    const bool zvalid = (zp - 2) >= z0;              // output z = zp-2 in chunk

#pragma unroll
    for (int f = 0; f < 5; ++f) {
      v8f acc = {0.f, 0.f, 0.f, 0.f, 0.f, 0.f, 0.f, 0.f};
      if (pvalid) {
#pragma unroll
        for (int c = 0; c < 5; ++c) {
          const int k0 = 4 * c + (half ? 2 : 0);     // B layout mirrors A
          v2f bv;
          bv.x = sTx[f * 320 + k0 * 16 + mn];
          bv.y = sTx[f * 320 + (k0 + 1) * 16 + mn];
          acc = __builtin_amdgcn_wmma_f32_16x16x4_f32(
              /*neg_a=*/false, Areg[c], /*neg_b=*/false, bv,
              /*c_mod=*/(short)0, acc, /*reuse_a=*/false, /*reuse_b=*/false);
        }
      }
      const int rbO = f * 1536 + so * 256 + lane;    // read base (P(zp-5))
      const int rbI = f * 1536 + si * 256 + lane;    // write base (P(zp))
#pragma unroll
      for (int j = 0; j < 8; ++j) {
        const float pv = acc[j];
        zs[f][j] += pv - sRing[rbO + j * 32];        // unconditional: warm-up reads zeros
        sRing[rbI + j * 32] = pv;                    // lane-private slot
      }
    }

    si = (si + 1 == 6) ? 0 : si + 1;
    so = (so + 1 == 6) ? 0 : so + 1;

    if (zvalid) {
      const float inv = 1.0f / 125.0f;
#pragma unroll
      for (int j = 0; j < 8; ++j) {
        const float ms = zs[0][j] * inv;
        const float mt = zs[1][j] * inv;
        const float vs = zs[2][j] * inv - ms * ms;
        const float vt = zs[3][j] * inv - mt * mt;
        const float cr = zs[4][j] * inv - ms * mt;
        lacc += (cr * cr) / (vs * vt + 1e-5f);
      }
    }
  }

  // Wave reduction (wave32), one partial per task.
  for (int off = 16; off > 0; off >>= 1) lacc += __shfl_down(lacc, off, 32);
  if (lane == 0) partial[blockIdx.x] = lacc;
}

// Deterministic fixed-order final reduction + nan_to_num on the scalar loss.
__global__ __launch_bounds__(256)
void lncc_reduce_kernel(const float* __restrict__ partial, float* __restrict__ out) {
  __shared__ float red[256];
  const int t = threadIdx.x;
  float s = 0.0f;
  for (int i = t; i < NTASK; i += 256) s += partial[i];   // 2304/256 = 9 each
  red[t] = s;
  __syncthreads();
  for (int stride = 128; stride > 0; stride >>= 1) {
    if (t < stride) red[t] += red[t + stride];
    __syncthreads();
  }
  if (t == 0) {
    float L = 1.0f - red[0] / (float)kNVOX;
    if (isnan(L) || isinf(L)) L = 1.0f;   // jnp.nan_to_num(nan=1, +-inf=1)
    out[0] = L;
  }
}

extern "C" void kernel_launch(void* const* d_in, const int* in_sizes, int n_in,
                              void* d_out, int out_size, void* d_ws, size_t ws_size,
                              hipStream_t stream) {
  const float* src = (const float*)d_in[0];   // "source"
  const float* tgt = (const float*)d_in[1];   // "target"
  float* partial = (float*)d_ws;              // NTASK * 4 bytes = 9216 B
  lncc_tile_kernel<<<NTASK, 32, 0, stream>>>(src, tgt, partial);
  lncc_reduce_kernel<<<1, 256, 0, stream>>>(partial, (float*)d_out);
}